// AdaptiveAttention_69793218560322
// MI455X (gfx1250) — compile-verified
//
#include <hip/hip_runtime.h>
#include <math.h>

#define NNODES 100000
#define NEDGES 1000000
#define INF    128
#define NH     4
#define ND     32
#define HD     128           // NH*ND
#define NEG_SLOPE 0.2f

typedef __attribute__((ext_vector_type(2))) float v2f;
typedef __attribute__((ext_vector_type(8))) float v8f;

// ---------------------------------------------------------------------------
// feat = X (nrows x 128) @ W (128 x 128), fp32 WMMA 16x16x4.
// One wave per 16x16 output tile; 8 waves per block cover the 128 output cols.
// ---------------------------------------------------------------------------
__global__ void gat_gemm_wmma(const float* __restrict__ X,
                              const float* __restrict__ W,
                              float* __restrict__ Y) {
  const int row0 = blockIdx.x << 4;              // 16-row tile
  const int wv   = threadIdx.x >> 5;             // wave id 0..7 -> col tile
  const int lane = threadIdx.x & 31;
  const int m    = lane & 15;                    // M index of A row
  const int koff = (lane >> 4) << 1;             // 0 for lanes 0-15, 2 for 16-31
  const int col  = (wv << 4) + m;                // N index (B/C/D lane mapping)

  const float* __restrict__ xrow = X + (size_t)(row0 + m) * INF;
  v8f c = {};
  #pragma unroll 4
  for (int k = 0; k < INF; k += 4) {
    // A fragment: lane holds X[row0+m][k+koff .. k+koff+1] (8B-aligned)
    v2f a = *(const v2f*)(xrow + k + koff);
    // B fragment: lane holds W[k+koff][col], W[k+koff+1][col]
    v2f b;
    b.x = W[(size_t)(k + koff) * HD + col];
    b.y = W[(size_t)(k + koff + 1) * HD + col];
    c = __builtin_amdgcn_wmma_f32_16x16x4_f32(false, a, false, b,
                                              (short)0, c, false, false);
  }
  // D layout: VGPR i -> M = i (lanes 0-15) / i+8 (lanes 16-31), N = col
  const int rbase = row0 + ((lane >> 4) << 3);
  #pragma unroll
  for (int i = 0; i < 8; ++i)
    Y[(size_t)(rbase + i) * HD + col] = c[i];
}

// el[n,h] = dot(feat[n,h,:], al[h,:]);  er likewise. One thread per (n,h).
__global__ void gat_attn_dots(const float* __restrict__ feat,
                              const float* __restrict__ al,
                              const float* __restrict__ ar,
                              float* __restrict__ el,
                              float* __restrict__ er) {
  int t = blockIdx.x * blockDim.x + threadIdx.x;
  if (t >= NNODES * NH) return;
  int n = t >> 2, hh = t & 3;
  const float* f = feat + (size_t)n * HD + hh * ND;
  const float* a = al + hh * ND;
  const float* b = ar + hh * ND;
  float sl = 0.f, sr = 0.f;
  #pragma unroll
  for (int d = 0; d < ND; ++d) {
    float v = f[d];
    sl += v * a[d];
    sr += v * b[d];
  }
  el[t] = sl;
  er[t] = sr;
}

// out[n, h*32+d] = b0[h*32+d] + b1[h*32+d]
__global__ void gat_init_out(float* __restrict__ out,
                             const float* __restrict__ b0,
                             const float* __restrict__ b1) {
  size_t t = (size_t)blockIdx.x * blockDim.x + threadIdx.x;
  if (t >= (size_t)NNODES * HD) return;
  int j = (int)(t & (HD - 1));
  out[t] = b0[j] + b1[j];
}

__global__ void gat_init_ms(float* __restrict__ mbuf, float* __restrict__ sbuf) {
  int t = blockIdx.x * blockDim.x + threadIdx.x;
  if (t >= NNODES * NH) return;
  mbuf[t] = -INFINITY;
  sbuf[t] = 0.f;
}

__device__ __forceinline__ float leaky(float v) {
  return v > 0.f ? v : NEG_SLOPE * v;
}

// Monotone-bits float atomic max (positives via signed max, negatives via
// unsigned min); correct with -inf initialization.
__device__ __forceinline__ void atomicMaxF32(float* addr, float v) {
  if (v >= 0.f)
    atomicMax((int*)addr, __float_as_int(v));
  else
    atomicMin((unsigned int*)addr, __float_as_uint(v));
}

// Pass 1: m[dst,h] = max over edges of leakyrelu(el[src]+er[dst])
__global__ void gat_edge_max(const int* __restrict__ src,
                             const int* __restrict__ dst,
                             const float* __restrict__ el,
                             const float* __restrict__ er,
                             float* __restrict__ mbuf) {
  int t = blockIdx.x * blockDim.x + threadIdx.x;
  if (t >= NEDGES * NH) return;
  int e = t >> 2, hh = t & 3;
  int sN = src[e], dN = dst[e];
  float v = leaky(el[sN * NH + hh] + er[dN * NH + hh]);
  atomicMaxF32(&mbuf[dN * NH + hh], v);
}

// Empty segments: -inf -> 0 (allow_zero_in_degree)
__global__ void gat_fix_m(float* __restrict__ mbuf) {
  int t = blockIdx.x * blockDim.x + threadIdx.x;
  if (t >= NNODES * NH) return;
  float m = mbuf[t];
  if (!isfinite(m)) mbuf[t] = 0.f;
}

// Pass 2: s[dst,h] += exp(e - m[dst,h])
__global__ void gat_edge_sum(const int* __restrict__ src,
                             const int* __restrict__ dst,
                             const float* __restrict__ el,
                             const float* __restrict__ er,
                             const float* __restrict__ mbuf,
                             float* __restrict__ sbuf) {
  int t = blockIdx.x * blockDim.x + threadIdx.x;
  if (t >= NEDGES * NH) return;
  int e = t >> 2, hh = t & 3;
  int sN = src[e], dN = dst[e];
  float v = leaky(el[sN * NH + hh] + er[dN * NH + hh]);
  float ex = __expf(v - mbuf[dN * NH + hh]);
  atomicAdd(&sbuf[dN * NH + hh], ex);
}

// Pass 3: out[dst,h,:] += feat[src,h,:] * alpha. One wave per edge, lane = d.
__global__ void gat_edge_agg(const int* __restrict__ src,
                             const int* __restrict__ dst,
                             const float* __restrict__ el,
                             const float* __restrict__ er,
                             const float* __restrict__ mbuf,
                             const float* __restrict__ sbuf,
                             const float* __restrict__ feat,
                             float* __restrict__ out) {
  int e = blockIdx.x * (blockDim.x >> 5) + (threadIdx.x >> 5);
  if (e >= NEDGES) return;
  int lane = threadIdx.x & 31;
  int sN = src[e], dN = dst[e];
  #pragma unroll
  for (int hh = 0; hh < NH; ++hh) {
    float v  = leaky(el[sN * NH + hh] + er[dN * NH + hh]);
    float ex = __expf(v - mbuf[dN * NH + hh]);
    float alpha = ex / fmaxf(sbuf[dN * NH + hh], 1e-20f);
    float val = feat[(size_t)sN * HD + hh * ND + lane] * alpha;
    atomicAdd(&out[(size_t)dN * HD + hh * ND + lane], val);
  }
}

extern "C" void kernel_launch(void* const* d_in, const int* in_sizes, int n_in,
                              void* d_out, int out_size, void* d_ws, size_t ws_size,
                              hipStream_t stream) {
  const float* h    = (const float*)d_in[0];
  const int*   srcs[2] = { (const int*)d_in[1], (const int*)d_in[3] };
  const int*   dsts[2] = { (const int*)d_in[2], (const int*)d_in[4] };
  const float* Ws[2]   = { (const float*)d_in[5],  (const float*)d_in[9]  };
  const float* als[2]  = { (const float*)d_in[6],  (const float*)d_in[10] };
  const float* ars[2]  = { (const float*)d_in[7],  (const float*)d_in[11] };
  const float* b0 = (const float*)d_in[8];
  const float* b1 = (const float*)d_in[12];
  float* out = (float*)d_out;

  // Workspace layout (reused across the two relations, processed serially)
  float* feat = (float*)d_ws;                       // N * 128
  float* el   = feat + (size_t)NNODES * HD;         // N * 4
  float* er   = el   + (size_t)NNODES * NH;
  float* mbuf = er   + (size_t)NNODES * NH;
  float* sbuf = mbuf + (size_t)NNODES * NH;

  const int B = 256;
  const int nNH   = (NNODES * NH + B - 1) / B;
  const int nOut  = (int)(((size_t)NNODES * HD + B - 1) / B);
  const int nEH   = (NEDGES * NH + B - 1) / B;
  const int nEdgeWaves = NEDGES / (B / 32);         // 8 waves/block

  gat_init_out<<<nOut, B, 0, stream>>>(out, b0, b1);

  for (int r = 0; r < 2; ++r) {
    gat_gemm_wmma<<<NNODES / 16, B, 0, stream>>>(h, Ws[r], feat);
    gat_attn_dots<<<nNH, B, 0, stream>>>(feat, als[r], ars[r], el, er);
    gat_init_ms<<<nNH, B, 0, stream>>>(mbuf, sbuf);
    gat_edge_max<<<nEH, B, 0, stream>>>(srcs[r], dsts[r], el, er, mbuf);
    gat_fix_m<<<nNH, B, 0, stream>>>(mbuf);
    gat_edge_sum<<<nEH, B, 0, stream>>>(srcs[r], dsts[r], el, er, mbuf, sbuf);
    gat_edge_agg<<<nEdgeWaves, B, 0, stream>>>(srcs[r], dsts[r], el, er,
                                               mbuf, sbuf, feat, out);
  }
}